// CachedMultiHeadAttentionDecoderSelf_68685116997773
// MI455X (gfx1250) — compile-verified
//
#include <hip/hip_runtime.h>
#include <math.h>

typedef float v2f __attribute__((ext_vector_type(2)));
typedef float v8f __attribute__((ext_vector_type(8)));

#define C_DIM   1024
#define B_DIM   16
#define T_DIM   8
#define H_DIM   16
#define HD      64
#define LCACHE  4096
#define TK_TOT  (LCACHE + T_DIM)            // 4104 keys total
#define NTILES  ((TK_TOT + 15) / 16)        // 257 key tiles of 16
#define SPLITS  16
#define TPB     ((NTILES + SPLITS - 1) / SPLITS)  // 17 tiles per split

// ---------------------------------------------------------------------------
// V_WMMA_F32_16X16X4_F32 : D(16x16,f32) = A(16x4,f32) x B(4x16,f32) + C
// A frag: a.x = A[m = lane%16][k = (lane<16 ? 0 : 2)], a.y = k+1
// B frag: b.x = B[k = (lane<16 ? 0 : 2)][n = lane%16], b.y = k+1
// C/D   : vgpr r -> (M = r + (lane<16 ? 0 : 8), N = lane%16)
// ---------------------------------------------------------------------------
__device__ __forceinline__ v8f wmma4(v2f a, v2f b, v8f c) {
  return __builtin_amdgcn_wmma_f32_16x16x4_f32(false, a, false, b, (short)0, c,
                                               false, false);
}

// ---------------------------------------------------------------------------
// CDNA5 async global->LDS copy (ASYNCcnt-tracked, no VGPR staging).
// VDST operand carries the per-lane wave-relative LDS byte offset
// (= low 32 bits of the generic pointer, per the LDS aperture layout).
// ---------------------------------------------------------------------------
__device__ __forceinline__ void async_ld_b128(const float* lds_dst,
                                              const float* gsrc) {
  const unsigned int       ldsoff = (unsigned int)(uintptr_t)lds_dst;
  const unsigned long long gaddr  = (unsigned long long)(uintptr_t)gsrc;
  asm volatile("global_load_async_to_lds_b128 %0, %1, off" ::"v"(ldsoff),
               "v"(gaddr)
               : "memory");
}
__device__ __forceinline__ void wait_async_le(int) {}
__device__ __forceinline__ void wait_async16() {
  asm volatile("s_wait_asynccnt 0x10" ::: "memory");
}
__device__ __forceinline__ void wait_async0() {
  asm volatile("s_wait_asynccnt 0x0" ::: "memory");
}

// ---------------------------------------------------------------------------
// One wave computes a 16x64 tile of Y = X (128x1024) * W^T (1024x1024) + bias
// ---------------------------------------------------------------------------
__device__ __forceinline__ void gemm_tile_16x64(const float* __restrict__ X,
                                                const float* __restrict__ W,
                                                const float* __restrict__ bias,
                                                float* __restrict__ Y,
                                                int mtile, int n0) {
  const int  lane = threadIdx.x & 31;
  const int  l16  = lane & 15;
  const bool lo   = lane < 16;
  const int  koff = lo ? 0 : 2;

  const float* xr = X + (size_t)(mtile * 16 + l16) * C_DIM;
  const float* wr[4];
#pragma unroll
  for (int nt = 0; nt < 4; ++nt)
    wr[nt] = W + (size_t)(n0 + nt * 16 + l16) * C_DIM;

  const v8f zero8 = {0.f, 0.f, 0.f, 0.f, 0.f, 0.f, 0.f, 0.f};
  v8f acc[4] = {zero8, zero8, zero8, zero8};

#pragma unroll 4
  for (int k = 0; k < C_DIM; k += 4) {
    v2f a = *(const v2f*)(xr + k + koff);
#pragma unroll
    for (int nt = 0; nt < 4; ++nt) {
      v2f b   = *(const v2f*)(wr[nt] + k + koff);
      acc[nt] = wmma4(a, b, acc[nt]);
    }
  }

#pragma unroll
  for (int nt = 0; nt < 4; ++nt) {
    const int   n  = n0 + nt * 16 + l16;
    const float bb = bias ? bias[n] : 0.f;
#pragma unroll
    for (int r = 0; r < 8; ++r) {
      const int row = mtile * 16 + r + (lo ? 0 : 8);
      Y[(size_t)row * C_DIM + n] = acc[nt][r] + bb;
    }
  }
}

// Kernel 1: q = x Wq^T + bq (-> ws), key = x Wk^T (-> out[1]), v = x Wv^T + bv
__global__ __launch_bounds__(32) void qkv_proj_kernel(
    const float* __restrict__ x, const float* __restrict__ Wq,
    const float* __restrict__ bq, const float* __restrict__ Wk,
    const float* __restrict__ Wv, const float* __restrict__ bv,
    float* __restrict__ qbuf, float* __restrict__ kout,
    float* __restrict__ vout) {
  const int    z    = blockIdx.z;
  const float* W    = (z == 0) ? Wq : (z == 1) ? Wk : Wv;
  const float* bias = (z == 0) ? bq : (z == 2) ? bv : nullptr;
  float*       Y    = (z == 0) ? qbuf : (z == 1) ? kout : vout;
  gemm_tile_16x64(x, W, bias, Y, blockIdx.y, blockIdx.x * 64);
}

// Kernel 4: out = wv Wo^T + bo
__global__ __launch_bounds__(32) void out_proj_kernel(
    const float* __restrict__ wvbuf, const float* __restrict__ Wo,
    const float* __restrict__ bo, float* __restrict__ out) {
  gemm_tile_16x64(wvbuf, Wo, bo, out, blockIdx.y, blockIdx.x * 64);
}

// ---------------------------------------------------------------------------
// Kernel 2: flash-decoding partials. One wave per (split, head, batch).
// K/V tiles double-buffered in LDS, filled with global_load_async_to_lds_b128
// (16 async ops per tile, wave-uniform), compute overlaps the next tile's
// HBM traffic behind s_wait_asynccnt 16.
// S^T = K_tile(16x64) x q^T(64x16) via 16 wmma; online softmax in-register;
// wv^T += V^T x P^T via 16 wmma per tile.
// ---------------------------------------------------------------------------
__global__ __launch_bounds__(32) void attn_partial_kernel(
    const float* __restrict__ kv, const float* __restrict__ qbuf,
    const float* __restrict__ knew, const float* __restrict__ vnew,
    float* __restrict__ pacc, float* __restrict__ pm,
    float* __restrict__ pl) {
  const int s = blockIdx.x, h = blockIdx.y, b = blockIdx.z;
  const int  lane = threadIdx.x & 31;
  const int  l16  = lane & 15;
  const bool lo   = lane < 16;
  const int  koff = lo ? 0 : 2;

  __shared__ float Kt[2][16][68];  // padded rows: conflict-free frag reads
  __shared__ float Vt[2][16][68];

  const float* kc_base =
      kv + ((size_t)(b * 2 + 1) * 2 + 0) * (size_t)LCACHE * C_DIM + h * HD;
  const float* vc_base =
      kv + ((size_t)(b * 2 + 1) * 2 + 1) * (size_t)LCACHE * C_DIM + h * HD;

  // q^T B-fragments, pre-scaled by hd^-0.5 (rows >= 8 zero padded)
  v2f qf[16];
#pragma unroll
  for (int ks = 0; ks < 16; ++ks) {
    if (l16 < T_DIM) {
      const float* qp =
          qbuf + (size_t)(b * T_DIM + l16) * C_DIM + h * HD + ks * 4 + koff;
      qf[ks].x = qp[0] * 0.125f;
      qf[ks].y = qp[1] * 0.125f;
    } else {
      qf[ks].x = 0.f;
      qf[ks].y = 0.f;
    }
  }

  const v8f zero8 = {0.f, 0.f, 0.f, 0.f, 0.f, 0.f, 0.f, 0.f};
  v8f   o[4]  = {zero8, zero8, zero8, zero8};  // wv^T accumulators
  float m_run = -__builtin_inff();
  float l_run = 0.f;

  int t0 = s * TPB;
  int t1 = t0 + TPB;
  if (t1 > NTILES) t1 = NTILES;

  const int hb = lane >> 4;  // which row of each pair this lane stages
  const int c4 = l16 * 4;

  // --- issue one tile's 16 async b128 loads (wave-uniform count) ---
  auto issue_tile = [&](int t, int buf) {
    const int rbase = t * 16;
    if (rbase < LCACHE) {  // tiles 0..255: entirely from the KV cache
#pragma unroll
      for (int i = 0; i < 8; ++i) {
        const int r = i * 2 + hb;
        const size_t go = (size_t)(rbase + r) * C_DIM + c4;
        async_ld_b128(&Kt[buf][r][c4], kc_base + go);
        async_ld_b128(&Vt[buf][r][c4], vc_base + go);
      }
    } else {  // tile 256: entirely from the freshly projected K/V rows
#pragma unroll
      for (int i = 0; i < 8; ++i) {
        const int r  = i * 2 + hb;
        int       lr = rbase + r - LCACHE;
        if (lr > T_DIM - 1) lr = T_DIM - 1;  // clamp: masked later
        const size_t go = (size_t)(b * T_DIM + lr) * C_DIM + h * HD + c4;
        async_ld_b128(&Kt[buf][r][c4], knew + go);
        async_ld_b128(&Vt[buf][r][c4], vnew + go);
      }
    }
  };

  issue_tile(t0, 0);

  for (int t = t0; t < t1; ++t) {
    const int buf = (t - t0) & 1;
    if (t + 1 < t1) {
      issue_tile(t + 1, buf ^ 1);  // prefetch next tile into other buffer
      wait_async16();              // current tile's 16 loads have landed
    } else {
      wait_async0();
    }

    // ---- S^T = K_tile x q^T : 16 wmma over hd ----
    v8f sacc = zero8;
#pragma unroll
    for (int ks = 0; ks < 16; ++ks) {
      v2f a;
      a.x  = Kt[buf][l16][ks * 4 + koff];
      a.y  = Kt[buf][l16][ks * 4 + koff + 1];
      sacc = wmma4(a, qf[ks], sacc);
    }

    // ---- online softmax (key dim in-register: 8/lane + half swap) ----
    const int kidx0 = t * 16 + (lo ? 0 : 8);
    float     p[8];
    float     tmax = -__builtin_inff();
#pragma unroll
    for (int r = 0; r < 8; ++r) {
      float sv = (kidx0 + r < TK_TOT) ? sacc[r] : -__builtin_inff();
      p[r]     = sv;
      tmax     = fmaxf(tmax, sv);
    }
    tmax              = fmaxf(tmax, __shfl_xor(tmax, 16, 32));
    const float m_new = fmaxf(m_run, tmax);
    const float alpha = __expf(m_run - m_new);
    float       lsum  = 0.f;
#pragma unroll
    for (int r = 0; r < 8; ++r) {
      const float e = __expf(p[r] - m_new);
      p[r]          = e;
      lsum += e;
    }
    l_run = l_run * alpha + lsum;
    m_run = m_new;
#pragma unroll
    for (int d = 0; d < 4; ++d) o[d] = o[d] * alpha;

    // ---- build P^T B-fragments (half-wave swaps) ----
    float psw[8];
#pragma unroll
    for (int r = 0; r < 8; ++r) psw[r] = __shfl_xor(p[r], 16, 32);
    v2f bf[4];
    bf[0].x = lo ? p[0] : psw[2];
    bf[0].y = lo ? p[1] : psw[3];
    bf[1].x = lo ? p[4] : psw[6];
    bf[1].y = lo ? p[5] : psw[7];
    bf[2].x = lo ? psw[0] : p[2];
    bf[2].y = lo ? psw[1] : p[3];
    bf[3].x = lo ? psw[4] : p[6];
    bf[3].y = lo ? psw[5] : p[7];

    // ---- wv^T += V^T x P^T : 16 wmma ----
#pragma unroll
    for (int ds_ = 0; ds_ < 4; ++ds_) {
#pragma unroll
      for (int kc4 = 0; kc4 < 4; ++kc4) {
        const int kc = kc4 * 4;
        v2f       a;
        a.x    = Vt[buf][kc + koff][ds_ * 16 + l16];
        a.y    = Vt[buf][kc + koff + 1][ds_ * 16 + l16];
        o[ds_] = wmma4(a, bf[kc4], o[ds_]);
      }
    }
  }

  // ---- store partials: acc (8q x 64d), m, l ----
  const float  l_tot = l_run + __shfl_xor(l_run, 16, 32);
  const size_t pbase = ((size_t)(b * H_DIM + h)) * SPLITS + s;
  if (lo && l16 < T_DIM) {
    pm[pbase * T_DIM + l16] = m_run;
    pl[pbase * T_DIM + l16] = l_tot;
  }
  if (l16 < T_DIM) {
#pragma unroll
    for (int ds_ = 0; ds_ < 4; ++ds_) {
#pragma unroll
      for (int r = 0; r < 8; ++r) {
        const int d = ds_ * 16 + r + (lo ? 0 : 8);
        pacc[pbase * (T_DIM * HD) + (size_t)l16 * HD + d] = o[ds_][r];
      }
    }
  }
}

// ---------------------------------------------------------------------------
// Kernel 3: combine split partials (log-sum-exp merge) -> wvbuf (B,T,C)
// ---------------------------------------------------------------------------
__global__ __launch_bounds__(64) void attn_combine_kernel(
    const float* __restrict__ pacc, const float* __restrict__ pm,
    const float* __restrict__ pl, float* __restrict__ wvbuf) {
  const int bh = blockIdx.x;
  const int b  = bh >> 4, h = bh & 15;
  const int d  = threadIdx.x;  // 0..63
  for (int t = 0; t < T_DIM; ++t) {
    float M = -__builtin_inff();
#pragma unroll
    for (int s = 0; s < SPLITS; ++s)
      M = fmaxf(M, pm[((size_t)bh * SPLITS + s) * T_DIM + t]);
    float L = 0.f, acc = 0.f;
#pragma unroll
    for (int s = 0; s < SPLITS; ++s) {
      const size_t ib = (size_t)bh * SPLITS + s;
      const float  w  = __expf(pm[ib * T_DIM + t] - M);
      L += pl[ib * T_DIM + t] * w;
      acc += pacc[ib * (T_DIM * HD) + (size_t)t * HD + d] * w;
    }
    wvbuf[(size_t)(b * T_DIM + t) * C_DIM + h * HD + d] = acc / L;
  }
}

// ---------------------------------------------------------------------------
extern "C" void kernel_launch(void* const* d_in, const int* in_sizes, int n_in,
                              void* d_out, int out_size, void* d_ws,
                              size_t ws_size, hipStream_t stream) {
  const float* x  = (const float*)d_in[0];
  const float* kv = (const float*)d_in[1];
  const float* Wq = (const float*)d_in[2];
  const float* bq = (const float*)d_in[3];
  const float* Wk = (const float*)d_in[4];
  const float* Wv = (const float*)d_in[5];
  const float* bv = (const float*)d_in[6];
  const float* Wo = (const float*)d_in[7];
  const float* bo = (const float*)d_in[8];

  float* out  = (float*)d_out;                 // (B,T,C)  131072 floats
  float* kout = out + 131072;                  // key output
  float* vout = out + 262144;                  // value output

  float* ws    = (float*)d_ws;
  float* qbuf  = ws;                            // 131072 floats
  float* wvbuf = ws + 131072;                   // 131072 floats
  float* pacc  = ws + 262144;                   // 256*16*512 = 2097152 floats
  float* pm    = ws + 262144 + 2097152;         // 32768 floats
  float* pl    = pm + 32768;                    // 32768 floats

  // 1) QKV projections (WMMA f32 GEMMs)
  qkv_proj_kernel<<<dim3(16, 8, 3), 32, 0, stream>>>(x, Wq, bq, Wk, Wv, bv,
                                                     qbuf, kout, vout);
  // 2) flash-decoding partials over the 4104-key cache (async LDS streaming)
  attn_partial_kernel<<<dim3(SPLITS, H_DIM, B_DIM), 32, 0, stream>>>(
      kv, qbuf, kout, vout, pacc, pm, pl);
  // 3) merge splits
  attn_combine_kernel<<<dim3(B_DIM * H_DIM), 64, 0, stream>>>(pacc, pm, pl,
                                                              wvbuf);
  // 4) output projection
  out_proj_kernel<<<dim3(16, 8), 32, 0, stream>>>(wvbuf, Wo, bo, out);
}